// QuantizerBlock_82884278879020
// MI455X (gfx1250) — compile-verified
//
#include <hip/hip_runtime.h>

// CDNA5 / gfx1250 vector-quantizer:
//   dists via V_WMMA_F32_16X16X4_F32 (exact f32 matmul on the matrix pipe),
//   A-tile preloaded as one load clause (max MLP), B fragments cache-hot,
//   argmin via wave32 xor-shuffle reduction, coalesced onehot+residual stores.

typedef __attribute__((ext_vector_type(2))) float v2f;
typedef __attribute__((ext_vector_type(8))) float v8f;

#define NUM_CODES 16
#define CODE_DIM  64

__global__ __launch_bounds__(256) void vq_wmma_kernel(
    const float* __restrict__ X,     // [N, 64]
    const float* __restrict__ CB,    // [16, 64]
    float* __restrict__ onehot,      // [N, 16]
    float* __restrict__ residual,    // [N, 64]
    int N)
{
    __shared__ int sidx[8][16];               // winning code per row, per wave

    const int lane = threadIdx.x & 31;        // wave32
    const int w    = threadIdx.x >> 5;        // wave id in block (0..7)
    const int tile = blockIdx.x * 8 + w;      // one 16-row tile per wave
    const int row0 = tile * 16;
    if (row0 >= N) return;

    const int n = lane & 15;                  // code column / A row for this lane
    const int h = lane >> 4;                  // K-half selector (0 or 1)

    // --- issue ALL A-fragment loads first: one clause, full memory-level
    //     parallelism while the cache-hot codebook work proceeds below.
    //     A layout (16x4 f32): lane holds row m=l&15; v0/v1 = K 2h, 2h+1.
    const float* xrow = X + (size_t)(row0 + n) * CODE_DIM;
    v2f afrag[16];
    #pragma unroll
    for (int t = 0; t < 16; ++t) {
        const float* p = xrow + 4 * t + 2 * h;
        v2f a; a.x = p[0]; a.y = p[1];
        afrag[t] = a;
    }

    // --- per-lane codebook norm ||c_n||^2 (codebook is 4KB, cache-hot) ---
    float cn = 0.f;
    const float4* cbn4 = reinterpret_cast<const float4*>(CB + n * CODE_DIM);
    #pragma unroll
    for (int t = 0; t < 16; ++t) {
        float4 c = cbn4[t];
        cn += c.x * c.x + c.y * c.y + c.z * c.z + c.w * c.w;
    }

    // --- B fragments: b[t] = (CB[n][4t+2h], CB[n][4t+2h+1]) ---
    v2f bfrag[16];
    #pragma unroll
    for (int t = 0; t < 16; ++t) {
        const float* p = CB + n * CODE_DIM + 4 * t + 2 * h;
        v2f b; b.x = p[0]; b.y = p[1];
        bfrag[t] = b;
    }

    // --- dot[m][n] = X[m,:]·CB[n,:] via 16 chained WMMAs (K=4 each) ---
    v8f acc = {0.f, 0.f, 0.f, 0.f, 0.f, 0.f, 0.f, 0.f};
    #pragma unroll
    for (int t = 0; t < 16; ++t) {
        acc = __builtin_amdgcn_wmma_f32_16x16x4_f32(
            /*neg_a=*/false, afrag[t], /*neg_b=*/false, bfrag[t],
            /*c_mod=*/(short)0, acc, /*reuse_a=*/false, /*reuse_b=*/false);
    }

    // --- per-row argmin over the 16 codes + coalesced one-hot stores ---
    // C/D layout: lane l holds column n=l&15; VGPR r holds row m = r + 8*(l>>4).
    #pragma unroll
    for (int r = 0; r < 8; ++r) {
        float s  = cn - 2.0f * acc[r];        // ||c||^2 - 2 x.c  (||x||^2 const per row)
        int   bi = n;
        #pragma unroll
        for (int off = 8; off >= 1; off >>= 1) {   // stays within each 16-lane half
            float so = __shfl_xor(s,  off, 32);
            int   bo = __shfl_xor(bi, off, 32);
            if (so < s || (so == s && bo < bi)) { s = so; bi = bo; }
        }
        const int    m = r + 8 * h;
        const size_t g = (size_t)row0 + (size_t)m;
        onehot[g * NUM_CODES + n] = (n == bi) ? 1.0f : 0.0f;  // 64B per half-wave
        if (lane == 0)  sidx[w][r]     = bi;
        if (lane == 16) sidx[w][8 + r] = bi;
    }
    __syncthreads();

    // --- residual: fully-coalesced float4 rows (256B per half-wave) ---
    #pragma unroll
    for (int it = 0; it < 8; ++it) {
        const int    m  = 2 * it + h;
        const size_t g  = (size_t)row0 + (size_t)m;
        const int    ci = sidx[w][m];
        const int    k4 = lane & 15;
        float4 xv = reinterpret_cast<const float4*>(X  + g  * CODE_DIM)[k4];
        float4 cv = reinterpret_cast<const float4*>(CB + (size_t)ci * CODE_DIM)[k4];
        float4 rv;
        rv.x = xv.x - cv.x; rv.y = xv.y - cv.y;
        rv.z = xv.z - cv.z; rv.w = xv.w - cv.w;
        reinterpret_cast<float4*>(residual + g * CODE_DIM)[k4] = rv;
    }
}

extern "C" void kernel_launch(void* const* d_in, const int* in_sizes, int n_in,
                              void* d_out, int out_size, void* d_ws, size_t ws_size,
                              hipStream_t stream) {
    const float* X  = (const float*)d_in[0];   // [N, 64]
    const float* CB = (const float*)d_in[1];   // [16, 64]
    const int N = in_sizes[0] / CODE_DIM;

    float* onehot   = (float*)d_out;                         // [N, 16] first
    float* residual = (float*)d_out + (size_t)N * NUM_CODES; // then [N, 64]

    const int tiles  = (N + 15) / 16;
    const int blocks = (tiles + 7) / 8;        // 8 waves (tiles) per 256-thread block
    vq_wmma_kernel<<<blocks, 256, 0, stream>>>(X, CB, onehot, residual, N);
}